// MBTR_30837865185354
// MI455X (gfx1250) — compile-verified
//
#include <hip/hip_runtime.h>
#include <math.h>

typedef __attribute__((ext_vector_type(2))) float v2f;
typedef __attribute__((ext_vector_type(8))) float v8f;

#define B_   32
#define NA   48
#define NE   4
#define NG   128
#define NP   (NA * (NA - 1))          // 2256 ordered pairs, divisible by 4
#define SIGMA_       0.05f
#define INV_SQRT2PI_ 0.39894228040143267794f
// W_SCALE == 1.0f (folded into the math below)

// ---------------------------------------------------------------------------
// Forward: mbtr[b, e(=e1*4+e2), g] = sum_p [seg[p]==e] * wf[p] * gauss(g; gf[p])
// One block (256 thr = 8 waves) per batch. Stage per-pair {gf,wf}/seg in LDS,
// then each wave owns one 16-wide grid tile and chains V_WMMA_F32_16X16X4_F32
// over the 2256-long pair (K) dimension in chunks of 4.
// ---------------------------------------------------------------------------
__global__ __launch_bounds__(256) void mbtr_fwd_kernel(
    const float* __restrict__ r, const int* __restrict__ z,
    const float* __restrict__ grid, float* __restrict__ out)
{
    __shared__ float s_r[NA * 3];
    __shared__ int   s_z[NA];
    __shared__ v2f   s_gw[NP];     // {gf, wf} packed -> single ds_load_b64
    __shared__ int   s_seg[NP];

    const int b   = blockIdx.x;
    const int tid = threadIdx.x;

    for (int i = tid; i < NA * 3; i += 256) s_r[i] = r[b * NA * 3 + i];
    for (int i = tid; i < NA;     i += 256) s_z[i] = z[i];
    __syncthreads();

    // per-pair precompute (cooperative)
    for (int p = tid; p < NP; p += 256) {
        int i = p / (NA - 1);
        int t = p - i * (NA - 1);
        int j = t + (t >= i);
        float dx = s_r[i * 3 + 0] - s_r[j * 3 + 0];
        float dy = s_r[i * 3 + 1] - s_r[j * 3 + 1];
        float dz = s_r[i * 3 + 2] - s_r[j * 3 + 2];
        float d  = sqrtf(dx * dx + dy * dy + dz * dz);
        v2f gw;
        gw.x = 1.0f / d;                 // gf
        gw.y = __expf(-d);               // wf (W_SCALE = 1)
        s_gw[p]  = gw;
        s_seg[p] = s_z[i] * NE + s_z[j];
    }
    __syncthreads();

    const int   lane  = tid & 31;
    const int   wave  = tid >> 5;              // 8 waves -> 8 grid tiles of 16
    const int   ncol  = wave * 16 + (lane & 15);
    const float gval  = grid[ncol];
    const float dxg   = grid[1] - grid[0];
    const float norm  = dxg * INV_SQRT2PI_ / SIGMA_;
    const float invs  = 1.0f / SIGMA_;
    const int   khalf = (lane < 16) ? 0 : 2;   // K half owned by this lane
    const int   m     = lane & 15;             // A-matrix row = bucket index

    v8f acc = {};
    for (int p = 0; p < NP; p += 4) {
        int k0 = p + khalf;
        // Unconditional LDS loads; the asm barrier keeps the compiler from
        // sinking them under an EXEC-masked if-region (we want v_cndmask,
        // not v_cmpx/s_and_saveexec, around the WMMA chain).
        v2f gw0 = s_gw[k0];
        v2f gw1 = s_gw[k0 + 1];
        int seg0 = s_seg[k0];
        int seg1 = s_seg[k0 + 1];
        float g0 = gw0.x, w0 = gw0.y;
        float g1 = gw1.x, w1 = gw1.y;
        asm volatile("" : "+v"(w0), "+v"(w1), "+v"(g0), "+v"(g1));
        // A fragment: A[e, k] = (seg[k]==e) ? wf[k] : 0   (plain select)
        float a0 = (seg0 == m) ? w0 : 0.0f;
        float a1 = (seg1 == m) ? w1 : 0.0f;
        // B fragment: B[k, n] = normalized gaussian at grid[n], center gf[k]
        float t0 = (gval - g0) * invs;
        float t1 = (gval - g1) * invs;
        float b0 = __expf(-0.5f * t0 * t0) * norm;
        float b1 = __expf(-0.5f * t1 * t1) * norm;
        v2f Af = {a0, a1};
        v2f Bf = {b0, b1};
        acc = __builtin_amdgcn_wmma_f32_16x16x4_f32(
            /*neg_a=*/false, Af, /*neg_b=*/false, Bf,
            /*c_mod=*/(short)0, acc, /*reuse_a=*/false, /*reuse_b=*/false);
    }

    // D layout: VGPR i -> M=i (lanes 0-15) / M=8+i (lanes 16-31), N = lane%16
    const int mbase = (lane < 16) ? 0 : 8;
#pragma unroll
    for (int i2 = 0; i2 < 8; ++i2) {
        int e = mbase + i2;                    // bucket e1*4+e2
        out[(b * 16 + e) * NG + ncol] = acc[i2];
    }
}

// ---------------------------------------------------------------------------
// Backward: one thread per (b, g, a). For each partner k:
//   coef = -wf * (gv + gf^2 * grid_div); contribution coef * u_c goes to
//   bucket (z[a], z[k]) [pair (a,k), side i] AND bucket (z[k], z[a])
//   [pair (k,a), side j; sign flips cancel]. Accumulate only S[e2][c],
//   then expand to the dense 16 buckets deterministically. All outputs
//   written (zeros included) -> no init pass, no atomics.
// ---------------------------------------------------------------------------
__global__ __launch_bounds__(256) void mbtr_div_kernel(
    const float* __restrict__ r, const int* __restrict__ z,
    const float* __restrict__ grid, float* __restrict__ outd)
{
    __shared__ float s_r[NA * 3];
    __shared__ int   s_z[NA];

    const int b   = blockIdx.x / 24;           // 24 blocks of 256 per batch
    const int tid = threadIdx.x;

    for (int i = tid; i < NA * 3; i += 256) s_r[i] = r[b * NA * 3 + i];
    for (int i = tid; i < NA;     i += 256) s_z[i] = z[i];
    __syncthreads();

    const int idx = (blockIdx.x % 24) * 256 + tid;  // 0..6143
    const int a   = idx % NA;
    const int g   = idx / NA;                       // 0..127

    const float gval   = grid[g];
    const float dxg    = grid[1] - grid[0];
    const float norm   = dxg * INV_SQRT2PI_ / SIGMA_;
    const float invs   = 1.0f / SIGMA_;
    const float inv_s2 = invs * invs;

    const float rax = s_r[a * 3 + 0];
    const float ray = s_r[a * 3 + 1];
    const float raz = s_r[a * 3 + 2];
    const int   za  = s_z[a];

    float S[NE][3];
#pragma unroll
    for (int e = 0; e < NE; ++e) { S[e][0] = 0.f; S[e][1] = 0.f; S[e][2] = 0.f; }

    for (int k = 0; k < NA; ++k) {
        if (k == a) continue;
        float dx = rax - s_r[k * 3 + 0];           // diff = r[i] - r[j], i=a
        float dy = ray - s_r[k * 3 + 1];
        float dz = raz - s_r[k * 3 + 2];
        float d  = sqrtf(dx * dx + dy * dy + dz * dz);
        float gf = 1.0f / d;
        float wf = __expf(-d);
        float t  = (gval - gf) * invs;
        float gv = __expf(-0.5f * t * t) * norm;
        float gdiv = gv * (gval - gf) * inv_s2;    // d gv / d gf
        float coef = -wf * (gv + gf * gf * gdiv);  // W_SCALE = 1
        float cu = coef * gf;                      // coef * (1/d)
        float cx = cu * dx, cy = cu * dy, cz = cu * dz;
        int   e2 = s_z[k];
#pragma unroll
        for (int e = 0; e < NE; ++e) {
            bool msk = (e2 == e);
            S[e][0] += msk ? cx : 0.0f;
            S[e][1] += msk ? cy : 0.0f;
            S[e][2] += msk ? cz : 0.0f;
        }
    }

    // dense expansion: out[e1][e2][c] = (e1==za)*S[e2][c] + (e2==za)*S[e1][c]
#pragma unroll
    for (int e1 = 0; e1 < NE; ++e1) {
#pragma unroll
        for (int e2 = 0; e2 < NE; ++e2) {
            float o0 = (e1 == za ? S[e2][0] : 0.f) + (e2 == za ? S[e1][0] : 0.f);
            float o1 = (e1 == za ? S[e2][1] : 0.f) + (e2 == za ? S[e1][1] : 0.f);
            float o2 = (e1 == za ? S[e2][2] : 0.f) + (e2 == za ? S[e1][2] : 0.f);
            size_t off = ((((size_t)((b * NE + e1) * NE + e2)) * NG + g) * NA + a) * 3;
            outd[off + 0] = o0;
            outd[off + 1] = o1;
            outd[off + 2] = o2;
        }
    }
}

extern "C" void kernel_launch(void* const* d_in, const int* in_sizes, int n_in,
                              void* d_out, int out_size, void* d_ws, size_t ws_size,
                              hipStream_t stream) {
    const float* r    = (const float*)d_in[0];   // (32,48,3) f32
    const int*   z    = (const int*)d_in[1];     // (48,) i32
    const float* grid = (const float*)d_in[2];   // (128,) f32
    float* out  = (float*)d_out;                 // mbtr: 32*4*4*128 = 65536
    float* outd = out + (size_t)B_ * NE * NE * NG;  // mbtr_div: 9437184

    mbtr_fwd_kernel<<<B_, 256, 0, stream>>>(r, z, grid, out);
    mbtr_div_kernel<<<(B_ * NG * NA) / 256, 256, 0, stream>>>(r, z, grid, outd);
}